// ShiftConv2d1_81028853006362
// MI455X (gfx1250) — compile-verified
//
#include <hip/hip_runtime.h>

// CDNA5 / gfx1250: wave32, WMMA fp32 16x16x4.
typedef __attribute__((ext_vector_type(2))) float v2f;
typedef __attribute__((ext_vector_type(8))) float v8f;

#define B_   4
#define C_   64
#define H_   512
#define W_   512
#define HW_  (H_ * W_)

// out[b,o,h,w] = sum_c W[o,c] * x[b,c,h-1,w] + bias[o]   (h==0 row -> bias only)
// GEMM view: M=64 (o), K=64 (c), N = b*h*w.  Wave tile: 64(M) x 16(N).
__global__ __launch_bounds__(256) void shiftconv1x1_wmma_f32(
    const float* __restrict__ x,      // [B, C, H, W]
    const float* __restrict__ wmat,   // [C_OUT, C_IN]
    const float* __restrict__ bias,   // [C_OUT]
    float* __restrict__ out)          // [B, C_OUT, H, W]
{
    __shared__ __align__(16) float w_lds[C_ * C_];   // 16 KB of 320 KB WGP LDS
    __shared__ __align__(16) float b_lds[C_];

    const int tid = threadIdx.x;

    // Cooperative weight preload: 4096 floats as 1024 float4 (b128 loads).
    for (int i = tid; i < (C_ * C_) / 4; i += 256)
        ((float4*)w_lds)[i] = ((const float4*)wmat)[i];
    if (tid < C_) b_lds[tid] = bias[tid];
    __syncthreads();

    const int wave = tid >> 5;
    const int lane = tid & 31;
    const int n    = lane & 15;   // N position within 16-wide tile
    const int half = lane >> 4;   // 0: lanes 0-15, 1: lanes 16-31

    // blockIdx.x = ((b * H) + h) * (W/128) + wchunk ; wave owns a 16-wide slice
    int blk = blockIdx.x;
    const int wc = blk & 3;          blk >>= 2;   // W/128 = 4 chunks
    const int h  = blk & (H_ - 1);   blk >>= 9;
    const int b  = blk;                            // 0..3
    const int w0 = wc * 128 + wave * 16;

    float* ob = out + (size_t)b * C_ * HW_ + (size_t)h * W_ + (w0 + n);

    // h == 0: shifted input is identically zero -> output is pure bias.
    // Block-uniform condition, so no EXEC divergence; skip GEMM + x loads.
    if (h == 0) {
        #pragma unroll
        for (int m = 0; m < 4; ++m) {
            const int o0 = 16 * m + 8 * half;
            const float4 bl0 = *(const float4*)&b_lds[o0];
            const float4 bl1 = *(const float4*)&b_lds[o0 + 4];
            ob[(size_t)(o0 + 0) * HW_] = bl0.x;
            ob[(size_t)(o0 + 1) * HW_] = bl0.y;
            ob[(size_t)(o0 + 2) * HW_] = bl0.z;
            ob[(size_t)(o0 + 3) * HW_] = bl0.w;
            ob[(size_t)(o0 + 4) * HW_] = bl1.x;
            ob[(size_t)(o0 + 5) * HW_] = bl1.y;
            ob[(size_t)(o0 + 6) * HW_] = bl1.z;
            ob[(size_t)(o0 + 7) * HW_] = bl1.w;
        }
        return;
    }

    const float* xb = x + (size_t)b * C_ * HW_ + (size_t)(h - 1) * W_ + (w0 + n);

    v8f acc0 = {}, acc1 = {}, acc2 = {}, acc3 = {};   // 4 chains: o-tiles 0..3

    // K reduction: 16 branch-free steps of K=4.  Per ISA 7.12.2 (16x4 fp32 A):
    //   lanes 0-15 hold K={0,1}, lanes 16-31 hold K={2,3}  -> pair {2*half, 2*half+1}
    // B/C/D: upper half-lanes offset by half the rows (same pairing for B rows).
    #pragma unroll 8
    for (int kk = 0; kk < 16; ++kk) {
        const int c = 4 * kk + 2 * half;           // this half-wave's channel pair

        v2f bf;                                    // shifted-x fragment (4x16)
        bf[0] = xb[(size_t)c * HW_];               // 64B contiguous per half-wave
        bf[1] = xb[(size_t)(c + 1) * HW_];

        const int ab = n * C_ + c;                 // W[o0+n, c..c+1] : ds_load_b64
        v2f a0 = *(const v2f*)&w_lds[ab];
        v2f a1 = *(const v2f*)&w_lds[ab + 16 * C_];
        v2f a2 = *(const v2f*)&w_lds[ab + 32 * C_];
        v2f a3 = *(const v2f*)&w_lds[ab + 48 * C_];

        acc0 = __builtin_amdgcn_wmma_f32_16x16x4_f32(false, a0, false, bf, (short)0, acc0, false, false);
        acc1 = __builtin_amdgcn_wmma_f32_16x16x4_f32(false, a1, false, bf, (short)0, acc1, false, false);
        acc2 = __builtin_amdgcn_wmma_f32_16x16x4_f32(false, a2, false, bf, (short)0, acc2, false, false);
        acc3 = __builtin_amdgcn_wmma_f32_16x16x4_f32(false, a3, false, bf, (short)0, acc3, false, false);
    }

    // Epilogue: D layout (16x16 f32 C/D): VGPR r -> row M = r + 8*half, col N = n.
    // o = 16*m + 8*half + r ; each b32 store writes 2 x 64B contiguous runs.
    const v8f accs[4] = {acc0, acc1, acc2, acc3};
    #pragma unroll
    for (int m = 0; m < 4; ++m) {
        const int o0 = 16 * m + 8 * half;
        const float4 bl0 = *(const float4*)&b_lds[o0];
        const float4 bl1 = *(const float4*)&b_lds[o0 + 4];
        ob[(size_t)(o0 + 0) * HW_] = accs[m][0] + bl0.x;
        ob[(size_t)(o0 + 1) * HW_] = accs[m][1] + bl0.y;
        ob[(size_t)(o0 + 2) * HW_] = accs[m][2] + bl0.z;
        ob[(size_t)(o0 + 3) * HW_] = accs[m][3] + bl0.w;
        ob[(size_t)(o0 + 4) * HW_] = accs[m][4] + bl1.x;
        ob[(size_t)(o0 + 5) * HW_] = accs[m][5] + bl1.y;
        ob[(size_t)(o0 + 6) * HW_] = accs[m][6] + bl1.z;
        ob[(size_t)(o0 + 7) * HW_] = accs[m][7] + bl1.w;
    }
}

extern "C" void kernel_launch(void* const* d_in, const int* in_sizes, int n_in,
                              void* d_out, int out_size, void* d_ws, size_t ws_size,
                              hipStream_t stream) {
    const float* x    = (const float*)d_in[0];   // [4,64,512,512] fp32
    const float* wm   = (const float*)d_in[1];   // [64,64] fp32
    const float* bias = (const float*)d_in[2];   // [64] fp32
    float* out        = (float*)d_out;           // [4,64,512,512] fp32

    const int grid = B_ * H_ * (W_ / 128);       // 8192 blocks x 256 threads (8 waves)
    shiftconv1x1_wmma_f32<<<grid, 256, 0, stream>>>(x, wm, bias, out);
}